// Splatter_65386582114828
// MI455X (gfx1250) — compile-verified
//
#include <hip/hip_runtime.h>
#include <hip/hip_bf16.h>
#include <math.h>

#define NG    2048
#define HPIX  128
#define WPIX  128
#define NEARZ 0.3f
#define EPSV  1e-4f

typedef __attribute__((ext_vector_type(2))) float v2f;
typedef __attribute__((ext_vector_type(8))) float v8f;

__device__ __forceinline__ float sigmoidf(float x) { return 1.0f / (1.0f + __expf(-x)); }

// ---------------------------------------------------------------------------
// Kernel 1: per-gaussian preprocessing.
// Produces: feat[i][8]  (gaussian feature vector for the rank-6 power GEMM)
//           rgbo[i][4]  (sigmoid rgb, opacity * (z>NEAR))
//           rr[i]       (camera-space distance, sort key)
// ---------------------------------------------------------------------------
__global__ __launch_bounds__(256) void splat_preprocess(
    const float* __restrict__ pos, const float* __restrict__ rgb,
    const float* __restrict__ opa, const float* __restrict__ quat,
    const float* __restrict__ scale, const float* __restrict__ rot,
    const float* __restrict__ tran,
    float* __restrict__ feat, float* __restrict__ rgbo, float* __restrict__ rr) {
  int i = blockIdx.x * blockDim.x + threadIdx.x;
  if (i >= NG) return;

  float R[9];
#pragma unroll
  for (int k = 0; k < 9; ++k) R[k] = rot[k];
  float p0 = pos[i * 3 + 0], p1 = pos[i * 3 + 1], p2 = pos[i * 3 + 2];
  float x = R[0] * p0 + R[1] * p1 + R[2] * p2 + tran[0];
  float y = R[3] * p0 + R[4] * p1 + R[5] * p2 + tran[1];
  float z = R[6] * p0 + R[7] * p1 + R[8] * p2 + tran[2];
  float r = sqrtf(x * x + y * y + z * z);
  float iz = 1.0f / z, iz2 = iz * iz, ir = 1.0f / r;
  float u = x * iz, v = y * iz;

  // quaternion -> rotation
  float qw = quat[i * 4 + 0], qx = quat[i * 4 + 1], qy = quat[i * 4 + 2], qz = quat[i * 4 + 3];
  float qn = rsqrtf(qw * qw + qx * qx + qy * qy + qz * qz);
  qw *= qn; qx *= qn; qy *= qn; qz *= qn;
  float Rm[9] = {
      1.0f - 2.0f * (qy * qy + qz * qz), 2.0f * (qx * qy - qw * qz), 2.0f * (qx * qz + qw * qy),
      2.0f * (qx * qy + qw * qz), 1.0f - 2.0f * (qx * qx + qz * qz), 2.0f * (qy * qz - qw * qx),
      2.0f * (qx * qz - qw * qy), 2.0f * (qy * qz + qw * qx), 1.0f - 2.0f * (qx * qx + qy * qy)};
  float s[3];
#pragma unroll
  for (int k = 0; k < 3; ++k) s[k] = fabsf(scale[i * 3 + k]) + 1e-4f;

  // cov3d = (Rm*diag(s)) (Rm*diag(s))^T
  float RS[9];
#pragma unroll
  for (int rI = 0; rI < 3; ++rI)
#pragma unroll
    for (int c = 0; c < 3; ++c) RS[rI * 3 + c] = Rm[rI * 3 + c] * s[c];
  float C[9];
#pragma unroll
  for (int rI = 0; rI < 3; ++rI)
#pragma unroll
    for (int c = 0; c < 3; ++c)
      C[rI * 3 + c] = RS[rI * 3 + 0] * RS[c * 3 + 0] + RS[rI * 3 + 1] * RS[c * 3 + 1] +
                      RS[rI * 3 + 2] * RS[c * 3 + 2];

  // JW rows 0,1 (only the 2x2 block of cov2d is needed).
  // J row0 = (1/z, 0, -x/z^2); row1 = (0, 1/z, -y/z^2). (row2 only feeds cov rows >=2)
  (void)ir;
  float JW0[3], JW1[3];
#pragma unroll
  for (int k = 0; k < 3; ++k) {
    JW0[k] = iz * R[0 * 3 + k] - x * iz2 * R[2 * 3 + k];
    JW1[k] = iz * R[1 * 3 + k] - y * iz2 * R[2 * 3 + k];
  }
  float M0[3], M1[3];
#pragma unroll
  for (int c = 0; c < 3; ++c) {
    M0[c] = JW0[0] * C[0 * 3 + c] + JW0[1] * C[1 * 3 + c] + JW0[2] * C[2 * 3 + c];
    M1[c] = JW1[0] * C[0 * 3 + c] + JW1[1] * C[1 * 3 + c] + JW1[2] * C[2 * 3 + c];
  }
  float a = M0[0] * JW0[0] + M0[1] * JW0[1] + M0[2] * JW0[2] + EPSV;
  float b = M0[0] * JW1[0] + M0[1] * JW1[1] + M0[2] * JW1[2];
  float c = M1[0] * JW1[0] + M1[1] * JW1[1] + M1[2] * JW1[2] + EPSV;
  float det = a * c - b * b;
  float idet = 1.0f / det;
  float ia = c * idet, ib = -b * idet, ic = a * idet;

  // power(pixel, gauss) = f(pixel) . g(gauss), f = [px^2, px*py, py^2, px, py, 1, 0, 0]
  feat[i * 8 + 0] = -0.5f * ia;
  feat[i * 8 + 1] = -ib;
  feat[i * 8 + 2] = -0.5f * ic;
  feat[i * 8 + 3] = ia * u + ib * v;
  feat[i * 8 + 4] = ib * u + ic * v;
  feat[i * 8 + 5] = -0.5f * (ia * u * u + 2.0f * ib * u * v + ic * v * v);
  feat[i * 8 + 6] = 0.0f;
  feat[i * 8 + 7] = 0.0f;

  float oe = (z > NEARZ) ? sigmoidf(opa[i]) : 0.0f;
  rgbo[i * 4 + 0] = sigmoidf(rgb[i * 3 + 0]);
  rgbo[i * 4 + 1] = sigmoidf(rgb[i * 3 + 1]);
  rgbo[i * 4 + 2] = sigmoidf(rgb[i * 3 + 2]);
  rgbo[i * 4 + 3] = oe;
  rr[i] = r;
}

// ---------------------------------------------------------------------------
// Kernel 2: bitonic argsort of 2048 depths in one workgroup, then gather
// sorted feature / rgbo tables.
// ---------------------------------------------------------------------------
__global__ __launch_bounds__(1024) void splat_sort_gather(
    const float* __restrict__ rr, const float* __restrict__ feat,
    const float* __restrict__ rgbo,
    float* __restrict__ sfeat, float* __restrict__ srgbo) {
  __shared__ float key[NG];
  __shared__ int   idx[NG];
  int t = threadIdx.x;
  for (int i = t; i < NG; i += 1024) { key[i] = rr[i]; idx[i] = i; }

  for (int k = 2; k <= NG; k <<= 1) {
    for (int j = k >> 1; j > 0; j >>= 1) {
      __syncthreads();
#pragma unroll
      for (int rep = 0; rep < 2; ++rep) {
        int i = t + rep * 1024;
        int ixj = i ^ j;
        if (ixj > i) {
          bool asc = ((i & k) == 0);
          float ki = key[i], kj = key[ixj];
          if ((ki > kj) == asc) {
            key[i] = kj; key[ixj] = ki;
            int tmp = idx[i]; idx[i] = idx[ixj]; idx[ixj] = tmp;
          }
        }
      }
    }
  }
  __syncthreads();
  for (int i = t; i < NG; i += 1024) {
    int o = idx[i];
#pragma unroll
    for (int q = 0; q < 8; ++q) sfeat[i * 8 + q] = feat[o * 8 + q];
    float4 vv = ((const float4*)rgbo)[o];
    ((float4*)srgbo)[i] = vv;
  }
}

// ---------------------------------------------------------------------------
// Kernel 3: render. Each wave owns 32 pixels (two 16-row WMMA M-tiles).
// Power tiles computed with V_WMMA_F32_16X16X4_F32 (rank-6 GEMM, K padded
// to 8 => two chained WMMAs per tile), round-tripped through padded LDS to
// transpose gaussian-major -> pixel-major, then sequential alpha blend.
// ---------------------------------------------------------------------------
__global__ __launch_bounds__(256) void splat_render(
    const float* __restrict__ sfeat, const float* __restrict__ srgbo,
    float* __restrict__ out) {
  __shared__ float ls_rgbo[NG * 4];     // 32 KB: sorted rgb+opacity table
  __shared__ float ls_scr[8][2 * 16 * 17];  // per-wave D-tile scratch, rows padded to 17

  // Stage rgb/opacity table once (block-wide, broadcast reads later).
  for (int i = threadIdx.x; i < NG; i += 256)
    ((float4*)ls_rgbo)[i] = ((const float4*)srgbo)[i];
  __syncthreads();

  const int wave = threadIdx.x >> 5;
  const int lane = threadIdx.x & 31;
  const int kgrp = lane >> 4;           // which K-half this lane carries (A/B frag)
  const int m    = lane & 15;           // row (A) / column (B) index within tile
  const int waveBase = blockIdx.x * 256 + wave * 32;

  // Constant A fragments for the wave's two pixel tiles.
  v2f a_lo[2], a_hi[2];
#pragma unroll
  for (int t = 0; t < 2; ++t) {
    int p = waveBase + t * 16 + m;
    int h = p >> 7, w = p & 127;
    float px = ((float)w - 64.0f + 0.5f) * (1.0f / 128.0f);
    float py = ((float)h - 64.0f + 0.5f) * (1.0f / 128.0f);
    float f0 = px * px, f1 = px * py, f2 = py * py, f3 = px, f4 = py, f5 = 1.0f;
    a_lo[t] = kgrp ? (v2f){f2, f3} : (v2f){f0, f1};  // K = {0,1} or {2,3}
    a_hi[t] = kgrp ? (v2f){0.0f, 0.0f} : (v2f){f4, f5};  // K = {4,5} or {6,7}(pad)
  }

  float* scr = &ls_scr[wave][0];
  const int ro = kgrp * 272 + m * 17;   // this lane's pixel row in scratch
  float T = 1.0f, accR = 0.0f, accG = 0.0f, accB = 0.0f;

  for (int base = 0; base < NG; base += 16) {
    // B fragment: gaussian column n = base+m, K rows selected by lane half.
    const float2* gp = (const float2*)(sfeat + (size_t)(base + m) * 8 + kgrp * 2);
    float2 gl = gp[0];     // K = kgrp*2 + {0,1}
    float2 gh = gp[2];     // K = kgrp*2 + {4,5}
    v2f b_lo = {gl.x, gl.y};
    v2f b_hi = {gh.x, gh.y};

    v8f d0 = {};
    v8f d1 = {};
    d0 = __builtin_amdgcn_wmma_f32_16x16x4_f32(false, a_lo[0], false, b_lo, (short)0, d0, false, false);
    d0 = __builtin_amdgcn_wmma_f32_16x16x4_f32(false, a_hi[0], false, b_hi, (short)0, d0, false, false);
    d1 = __builtin_amdgcn_wmma_f32_16x16x4_f32(false, a_lo[1], false, b_lo, (short)0, d1, false, false);
    d1 = __builtin_amdgcn_wmma_f32_16x16x4_f32(false, a_hi[1], false, b_hi, (short)0, d1, false, false);

    // Spill D tiles to padded LDS (gaussian-major -> pixel-major transpose).
#pragma unroll
    for (int j = 0; j < 8; ++j) {
      int row = j + kgrp * 8;
      scr[row * 17 + m] = d0[j];
      scr[272 + row * 17 + m] = d1[j];
    }
    __asm__ volatile("" ::: "memory");  // keep LDS store->load order (same wave, DS in-order)

    // Sequential front-to-back blend for this lane's pixel.
#pragma unroll
    for (int n = 0; n < 16; ++n) {
      float pw = scr[ro + n];
      float4 co = ((const float4*)ls_rgbo)[base + n];
      float alpha = fminf(co.w * __expf(pw), 0.99f);
      float wgt = T * alpha;
      accR += wgt * co.x;
      accG += wgt * co.y;
      accB += wgt * co.z;
      T *= 1.0f - alpha;
    }
    __asm__ volatile("" ::: "memory");  // reads done before next chunk overwrites
  }

  int p = waveBase + lane;
  out[p * 3 + 0] = accR;
  out[p * 3 + 1] = accG;
  out[p * 3 + 2] = accB;
}

extern "C" void kernel_launch(void* const* d_in, const int* in_sizes, int n_in,
                              void* d_out, int out_size, void* d_ws, size_t ws_size,
                              hipStream_t stream) {
  const float* pos   = (const float*)d_in[0];
  const float* rgb   = (const float*)d_in[1];
  const float* opa   = (const float*)d_in[2];
  const float* quat  = (const float*)d_in[3];
  const float* scale = (const float*)d_in[4];
  const float* rot   = (const float*)d_in[5];
  const float* tran  = (const float*)d_in[6];
  float* out = (float*)d_out;

  float* ws    = (float*)d_ws;
  float* feat  = ws;               // 2048*8
  float* rgbo  = ws + 16384;       // 2048*4
  float* rr    = ws + 24576;       // 2048
  float* sfeat = ws + 26624;       // 2048*8
  float* srgbo = ws + 43008;       // 2048*4  (total 51200 floats = 200 KB)

  splat_preprocess<<<NG / 256, 256, 0, stream>>>(pos, rgb, opa, quat, scale, rot, tran,
                                                 feat, rgbo, rr);
  splat_sort_gather<<<1, 1024, 0, stream>>>(rr, feat, rgbo, sfeat, srgbo);
  splat_render<<<(HPIX * WPIX) / 256, 256, 0, stream>>>(sfeat, srgbo, out);
}